// myModule_21586505630045
// MI455X (gfx1250) — compile-verified
//
#include <hip/hip_runtime.h>
#include <hip/hip_bf16.h>

// Problem constants (hardcoded in reference)
#define C_DIM 640
#define HH 5
#define WW 5
#define HW 25            // 5*5 pixels
#define KK 9             // 3x3 taps
#define NS 5             // support instances
#define NQ 75            // query instances
#define OKK (C_DIM * KK) // 5760 = C*K*K
#define CHW (C_DIM * HW) // 16000

typedef __attribute__((ext_vector_type(2))) float v2f;
typedef __attribute__((ext_vector_type(8))) float v8f;

__device__ __forceinline__ float sigmoidf_(float x) {
    return 1.0f / (1.0f + __expf(-x));
}

// t[c*25+p] = mean over 5 support instances of s[n, c, p]
__global__ void k_mean_support(const float* __restrict__ s, float* __restrict__ t) {
    int i = blockIdx.x * blockDim.x + threadIdx.x;
    if (i >= CHW) return;
    float acc = 0.f;
#pragma unroll
    for (int n = 0; n < NS; ++n) acc += s[n * CHW + i];
    t[i] = acc * (1.0f / NS);
}

// im2col for the 3x3 clm conv: Acol[p, c*9+ki*3+kj] = t[c, pix(h+ki-1, w+kj-1)] (zero-pad)
// rows 25..31 are zero padding so M is a multiple of 16.
__global__ void k_build_acol(const float* __restrict__ t, float* __restrict__ Acol) {
    int i = blockIdx.x * blockDim.x + threadIdx.x;
    if (i >= 32 * OKK) return;
    int row  = i / OKK;   // pixel (padded to 32)
    int kcol = i % OKK;   // c*9 + ki*3 + kj
    float v = 0.f;
    if (row < HW) {
        int c  = kcol / KK, r = kcol % KK;
        int ki = r / 3,     kj = r % 3;
        int h  = row / WW,  w  = row % WW;
        int hh = h + ki - 1, ww = w + kj - 1;
        if (hh >= 0 && hh < HH && ww >= 0 && ww < WW) v = t[c * HW + hh * WW + ww];
    }
    Acol[i] = v;
}

// Stack activations for the fused dk GEMM:
// rows 0..124  : support pixel vectors   X[n*25+p, c] = s[n, c, p]
// rows 125..149: clm output pixel vectors X[125+p, c] = clm[c*25+p]
// rows 150..159: zero padding (M multiple of 16)
__global__ void k_build_x(const float* __restrict__ s, const float* __restrict__ clm,
                          float* __restrict__ X) {
    int i = blockIdx.x * blockDim.x + threadIdx.x;
    if (i >= 160 * C_DIM) return;
    int row = i / C_DIM, c = i % C_DIM;
    float v = 0.f;
    if (row < NS * HW) {
        int n = row / HW, p = row % HW;
        v = s[n * CHW + c * HW + p];
    } else if (row < NS * HW + HW) {
        v = clm[c * HW + (row - NS * HW)];
    }
    X[i] = v;
}

// Generic fp32 WMMA GEMM:  D = act(A * B^T)
//   A: (Mtiles*16) x Kd, row-major, lda = Kd
//   B: N x Kd, row-major (so we compute A.B^T), ldb = Kd
// One wave per 16x16 tile; 8 waves per block cover 128 N-columns.
// act: 0 = ReLU, 1 = sigmoid.  transStore: store D[col*ldc+row] instead of D[row*ldc+col].
// f32 16x16x4 fragment layouts per CDNA5 ISA 7.12.2:
//   A: lane(0..15)=M, vgpr{0,1}=K{0,1}; lanes 16..31 hold K{2,3}.  B symmetric (lane=N).
//   C/D: 8 vgprs; lanes 0..15 -> M=r, lanes 16..31 -> M=r+8; lane%16 = N.
__global__ void k_gemm_wmma(const float* __restrict__ A, const float* __restrict__ B,
                            float* __restrict__ D, int Kd, int Mrows, int ldc,
                            int act, int transStore) {
    int lane    = threadIdx.x & 31;
    int wave    = threadIdx.x >> 5;
    int nTile   = blockIdx.x * 8 + wave;
    int mTile   = blockIdx.y;
    int lane_lo = lane & 15;
    int is_hi   = lane >> 4;

    const float* arow = A + (size_t)(mTile * 16 + lane_lo) * Kd + 2 * is_hi;
    const float* brow = B + (size_t)(nTile * 16 + lane_lo) * Kd + 2 * is_hi;

    v8f acc = {};
    for (int k = 0; k < Kd; k += 4) {
        v2f a = *(const v2f*)(arow + k);   // 8B coalesced fragment load
        v2f b = *(const v2f*)(brow + k);
        acc = __builtin_amdgcn_wmma_f32_16x16x4_f32(
            /*neg_a=*/false, a, /*neg_b=*/false, b,
            /*c_mod=*/(short)0, acc, /*reuse_a=*/false, /*reuse_b=*/false);
    }

    int col = nTile * 16 + lane_lo;
#pragma unroll
    for (int r = 0; r < 8; ++r) {
        int row = mTile * 16 + r + 8 * is_hi;
        if (row < Mrows) {
            float v = acc[r];
            v = act ? sigmoidf_(v) : fmaxf(v, 0.f);
            if (transStore) D[(size_t)col * ldc + row] = v;
            else            D[(size_t)row * ldc + col] = v;
        }
    }
}

// Support branch: adapted = s + mean_taps(unfold(s) * taskK * instK); out = spatial mean.
// G rows: [n*25+p] = instK rows, [125+p] = taskK rows; tap index = c*9 + ki*3 + kj.
__global__ void k_final_support(const float* __restrict__ s, const float* __restrict__ G,
                                float* __restrict__ out) {
    int i = blockIdx.x * blockDim.x + threadIdx.x;
    if (i >= NS * C_DIM) return;
    int n = i / C_DIM, c = i % C_DIM;
    const float* sp = s + n * CHW + c * HW;
    float acc = 0.f;
    for (int p = 0; p < HW; ++p) {
        int h = p / WW, w = p % WW;
        const float* gi = G + (size_t)(n * HW + p) * OKK + c * KK;
        const float* gt = G + (size_t)(NS * HW + p) * OKK + c * KK;
        float conv = 0.f;
#pragma unroll
        for (int ki = 0; ki < 3; ++ki)
#pragma unroll
            for (int kj = 0; kj < 3; ++kj) {
                int hh = h + ki - 1, ww2 = w + kj - 1;
                float xv = (hh >= 0 && hh < HH && ww2 >= 0 && ww2 < WW) ? sp[hh * WW + ww2] : 0.f;
                conv += xv * gi[ki * 3 + kj] * gt[ki * 3 + kj];
            }
        acc += sp[p] + conv * (1.0f / KK);
    }
    out[i] = acc * (1.0f / HW);
}

// Query branch: adapted = q + mean_taps(unfold(q) * taskK); out = spatial mean.
__global__ void k_final_query(const float* __restrict__ q, const float* __restrict__ G,
                              float* __restrict__ out) {
    int i = blockIdx.x * blockDim.x + threadIdx.x;
    if (i >= NQ * C_DIM) return;
    int n = i / C_DIM, c = i % C_DIM;
    const float* qp = q + n * CHW + c * HW;
    float acc = 0.f;
    for (int p = 0; p < HW; ++p) {
        int h = p / WW, w = p % WW;
        const float* gt = G + (size_t)(NS * HW + p) * OKK + c * KK;
        float conv = 0.f;
#pragma unroll
        for (int ki = 0; ki < 3; ++ki)
#pragma unroll
            for (int kj = 0; kj < 3; ++kj) {
                int hh = h + ki - 1, ww2 = w + kj - 1;
                float xv = (hh >= 0 && hh < HH && ww2 >= 0 && ww2 < WW) ? qp[hh * WW + ww2] : 0.f;
                conv += xv * gt[ki * 3 + kj];
            }
        acc += qp[p] + conv * (1.0f / KK);
    }
    out[NS * C_DIM + i] = acc * (1.0f / HW);
}

extern "C" void kernel_launch(void* const* d_in, const int* in_sizes, int n_in,
                              void* d_out, int out_size, void* d_ws, size_t ws_size,
                              hipStream_t stream) {
    (void)in_sizes; (void)n_in; (void)out_size; (void)ws_size;
    const float* s    = (const float*)d_in[0]; // (1,5,16000)  -> (5,640,5,5)
    const float* q    = (const float*)d_in[1]; // (1,75,16000) -> (75,640,5,5)
    const float* Wdk  = (const float*)d_in[2]; // (5760,640)
    const float* Wclm = (const float*)d_in[3]; // (640,640,3,3) == (640,5760)

    float* ws   = (float*)d_ws;
    float* t    = ws;                 // 16,000   mean support map (c,p)
    float* Acol = t    + CHW;         // 184,320  im2col 32 x 5760
    float* clmO = Acol + 32 * OKK;    // 16,000   relu(conv) as (c,p)
    float* X    = clmO + CHW;         // 102,400  stacked activations 160 x 640
    float* G    = X    + 160 * C_DIM; // 921,600  sigmoid GEMM out 160 x 5760
    float* out  = (float*)d_out;      // [5*640 support | 75*640 query]

    k_mean_support<<<(CHW + 255) / 256, 256, 0, stream>>>(s, t);
    k_build_acol<<<(32 * OKK + 255) / 256, 256, 0, stream>>>(t, Acol);

    // clm conv as GEMM: M=25(pad32), N=640, K=5760, ReLU, store transposed to (c,p)
    k_gemm_wmma<<<dim3(40 / 8, 2), 256, 0, stream>>>(Acol, Wclm, clmO,
                                                     /*Kd=*/OKK, /*Mrows=*/HW, /*ldc=*/HW,
                                                     /*act=*/0, /*transStore=*/1);

    k_build_x<<<(160 * C_DIM + 255) / 256, 256, 0, stream>>>(s, clmO, X);

    // fused dk_generator GEMM: M=150(pad160), N=5760, K=640, sigmoid
    k_gemm_wmma<<<dim3(360 / 8, 10), 256, 0, stream>>>(X, Wdk, G,
                                                       /*Kd=*/C_DIM, /*Mrows=*/150, /*ldc=*/OKK,
                                                       /*act=*/1, /*transStore=*/0);

    k_final_support<<<(NS * C_DIM + 255) / 256, 256, 0, stream>>>(s, G, out);
    k_final_query<<<(NQ * C_DIM + 255) / 256, 256, 0, stream>>>(q, G, out);
}